// FedG_DQN_3307124818437
// MI455X (gfx1250) — compile-verified
//
#include <hip/hip_runtime.h>
#include <hip/hip_bf16.h>

typedef __attribute__((ext_vector_type(2))) float v2f;
typedef __attribute__((ext_vector_type(8))) float v8f;

#define FEAT 128

#if __has_builtin(__builtin_amdgcn_global_load_async_to_lds_b128)
#define HAVE_ASYNC_LDS 1
typedef int v4i __attribute__((vector_size(4 * sizeof(int))));
typedef __attribute__((address_space(1))) v4i* gv4i_t;   // global (LangAS device)
typedef __attribute__((address_space(3))) v4i* lv4i_t;   // LDS
#endif

// ---------------------------------------------------------------------------
// Edge scatter: agg[dst] += feat[src], cnt[dst] += 1   (mean aggregation)
// One wave32 per edge: 32 lanes x float4 = 128 floats.
// Feature tables (51 MB) are L2-resident on MI455X (192 MB L2).
// ---------------------------------------------------------------------------
__global__ void scatter_accum_kernel(const float* __restrict__ feat,
                                     const int* __restrict__ src,
                                     const int* __restrict__ dst,
                                     float* __restrict__ agg,
                                     float* __restrict__ cnt,
                                     int nEdges, int addCnt)
{
    const int lane   = threadIdx.x & 31;
    const int gwave  = (blockIdx.x * blockDim.x + threadIdx.x) >> 5;
    const int nWaves = (gridDim.x * blockDim.x) >> 5;

    for (int e = gwave; e < nEdges; e += nWaves) {
        // prefetch edge indices a few iterations ahead (global_prefetch_b8)
        if (lane == 0 && (e + 4 * nWaves) < nEdges) {
            __builtin_prefetch(src + e + 4 * nWaves, 0, 1);
            __builtin_prefetch(dst + e + 4 * nWaves, 0, 1);
        }
        const int s = src[e];
        const int d = dst[e];
        const float4 v = ((const float4*)(feat + (size_t)s * FEAT))[lane];
        float* a = agg + (size_t)d * FEAT + lane * 4;
        atomicAdd(a + 0, v.x);
        atomicAdd(a + 1, v.y);
        atomicAdd(a + 2, v.z);
        atomicAdd(a + 3, v.w);
        if (addCnt && lane == 0) atomicAdd(cnt + d, 1.0f);
    }
}

// ---------------------------------------------------------------------------
// out = act( (agg/deg) @ Wl + self @ Wr + b ), fp32 WMMA 16x16x4.
//
// - Both full weight matrices live in LDS (128 KB -> 2 blocks/WGP of 320 KB),
//   staged via GLOBAL_LOAD_ASYNC_TO_LDS_B128 when the builtin exists.
// - One wave per 16-node row tile, EIGHT v8f accumulators = all 128 output
//   columns, so each A fragment is loaded from L2 exactly once.
// - Mean normalization folded into the A load (2 VALU muls per k-step).
//
// VGPR layouts per ISA:
//   A (16x4 f32): lane L holds A[m][k0+2*half], A[m][k0+2*half+1]
//                 (m=L&15, half=L>>4) -> one 8B v2f global load.
//   B (4x16 f32): lane L, VGPR j holds W[k0+2*half+j][n0+m] -> paired DS
//                 loads, conflict-free (consecutive columns across lanes).
//   D: VGPR v, lane L -> (M = v + 8*half, N = m).
// ---------------------------------------------------------------------------
__global__ void sage_layer_wmma(const float* __restrict__ aggFeat,  // raw sums
                                const float* __restrict__ cnt,      // degrees
                                const float* __restrict__ selfFeat,
                                const float* __restrict__ Wl,       // [128,128] (k,n)
                                const float* __restrict__ Wr,       // [128,128]
                                const float* __restrict__ bias,     // [128]
                                float* __restrict__ outFeat,        // [N,128]
                                int nNodes, int doRelu)
{
    __shared__ float sWl[FEAT * FEAT];
    __shared__ float sWr[FEAT * FEAT];

    const int tid  = threadIdx.x;
    const int lane = tid & 31;
    const int wv   = tid >> 5;
    const int m    = lane & 15;
    const int half = lane >> 4;
    const int nTiles        = nNodes >> 4;           // N divisible by 16
    const int wavesPerBlock = blockDim.x >> 5;
    const int gwave         = blockIdx.x * wavesPerBlock + wv;
    const int gstride       = gridDim.x * wavesPerBlock;

    // ---- stage both weight matrices into LDS ----
#ifdef HAVE_ASYNC_LDS
    {
        gv4i_t gl = (gv4i_t)(size_t)Wl;      // global addrspace via integer cast
        gv4i_t gr = (gv4i_t)(size_t)Wr;
        lv4i_t ll = (lv4i_t)sWl;             // generic -> LDS addrspacecast
        lv4i_t lr = (lv4i_t)sWr;
        const int nChunks = FEAT * FEAT / 4; // 16 B per chunk
        for (int i = tid; i < nChunks; i += blockDim.x) {
            __builtin_amdgcn_global_load_async_to_lds_b128(gl + i, ll + i, 0, 0);
            __builtin_amdgcn_global_load_async_to_lds_b128(gr + i, lr + i, 0, 0);
        }
        asm volatile("s_wait_asynccnt 0x0" ::: "memory");
    }
#else
    for (int i = tid; i < FEAT * FEAT; i += blockDim.x) {
        sWl[i] = Wl[i];
        sWr[i] = Wr[i];
    }
#endif
    __syncthreads();

    // bias per n-tile (independent of the row tile)
    float bv[8];
    #pragma unroll
    for (int nt = 0; nt < 8; ++nt) bv[nt] = bias[nt * 16 + m];

    for (int t = gwave; t < nTiles; t += gstride) {
        const float* Am = aggFeat  + (size_t)t * 16 * FEAT;
        const float* Ax = selfFeat + (size_t)t * 16 * FEAT;
        const float inv = 1.0f / fmaxf(cnt[t * 16 + m], 1.0f);

        v8f acc[8];
        #pragma unroll
        for (int nt = 0; nt < 8; ++nt) {
            #pragma unroll
            for (int v = 0; v < 8; ++v) acc[nt][v] = bv[nt];
        }

        #pragma unroll 2
        for (int kk = 0; kk < 32; ++kk) {
            const int ka = kk * 4 + 2 * half;
            v2f am = *(const v2f*)(Am + (size_t)m * FEAT + ka);
            v2f ax = *(const v2f*)(Ax + (size_t)m * FEAT + ka);
            am.x *= inv;                     // fold mean division into A
            am.y *= inv;
            #pragma unroll
            for (int nt = 0; nt < 8; ++nt) {
                const int n0 = nt * 16;
                v2f bl, br;
                bl.x = sWl[(ka + 0) * FEAT + n0 + m];
                bl.y = sWl[(ka + 1) * FEAT + n0 + m];
                br.x = sWr[(ka + 0) * FEAT + n0 + m];
                br.y = sWr[(ka + 1) * FEAT + n0 + m];
                acc[nt] = __builtin_amdgcn_wmma_f32_16x16x4_f32(
                              false, am, false, bl, (short)0, acc[nt], false, false);
                acc[nt] = __builtin_amdgcn_wmma_f32_16x16x4_f32(
                              false, ax, false, br, (short)0, acc[nt], false, false);
            }
        }

        #pragma unroll
        for (int nt = 0; nt < 8; ++nt) {
            float* orow = outFeat + (size_t)t * 16 * FEAT + nt * 16 + m;
            #pragma unroll
            for (int v = 0; v < 8; ++v) {
                float val = acc[nt][v];
                if (doRelu) val = fmaxf(val, 0.0f);
                orow[(size_t)(v + 8 * half) * FEAT] = val;
            }
        }
    }
}

// ---------------------------------------------------------------------------
// MLP head: cat = [curr_emb | dest_emb | nbr_emb] per neighbor (K x 384),
// hidden = relu(cat @ Wlin1 + blin1), q = hidden @ Wlin2 + blin2.
// Tiny (K=64); single block.
// ---------------------------------------------------------------------------
__global__ void head_kernel(const float* __restrict__ emb,
                            const int* __restrict__ currIdx,
                            const int* __restrict__ destIdx,
                            const int* __restrict__ nbrIdx,
                            const float* __restrict__ Wlin1, // [384,128]
                            const float* __restrict__ blin1, // [128]
                            const float* __restrict__ Wlin2, // [128,1]
                            const float* __restrict__ blin2, // [1]
                            float* __restrict__ qOut, int K)
{
    __shared__ float sCur[FEAT];
    __shared__ float sDst[FEAT];
    __shared__ float sHid[64 * FEAT];
    const int tid = threadIdx.x;

    if (tid < FEAT) {
        sCur[tid] = emb[(size_t)currIdx[0] * FEAT + tid];
        sDst[tid] = emb[(size_t)destIdx[0] * FEAT + tid];
    }
    __syncthreads();

    for (int idx = tid; idx < K * FEAT; idx += blockDim.x) {
        const int r = idx >> 7;
        const int j = idx & (FEAT - 1);
        const float* ne = emb + (size_t)nbrIdx[r] * FEAT;
        float s = blin1[j];
        for (int k = 0; k < FEAT; ++k) s += sCur[k] * Wlin1[k * FEAT + j];
        for (int k = 0; k < FEAT; ++k) s += sDst[k] * Wlin1[(FEAT + k) * FEAT + j];
        for (int k = 0; k < FEAT; ++k) s += ne[k]   * Wlin1[(2 * FEAT + k) * FEAT + j];
        sHid[idx] = fmaxf(s, 0.0f);
    }
    __syncthreads();

    for (int r = tid; r < K; r += blockDim.x) {
        float s = blin2[0];
        for (int j = 0; j < FEAT; ++j) s += sHid[r * FEAT + j] * Wlin2[j];
        qOut[r] = s;
    }
}

// ---------------------------------------------------------------------------
extern "C" void kernel_launch(void* const* d_in, const int* in_sizes, int n_in,
                              void* d_out, int out_size, void* d_ws, size_t ws_size,
                              hipStream_t stream)
{
    const float* x    = (const float*)d_in[0];
    const int*   ei   = (const int*)d_in[1];
    const int*   cur  = (const int*)d_in[2];
    const int*   dstI = (const int*)d_in[3];
    const int*   nbr  = (const int*)d_in[4];
    const float* W1l  = (const float*)d_in[5];
    const float* W1r  = (const float*)d_in[6];
    const float* b1   = (const float*)d_in[7];
    const float* W2l  = (const float*)d_in[8];
    const float* W2r  = (const float*)d_in[9];
    const float* b2   = (const float*)d_in[10];
    const float* Wl1  = (const float*)d_in[11];
    const float* bl1  = (const float*)d_in[12];
    const float* Wl2  = (const float*)d_in[13];
    const float* bl2  = (const float*)d_in[14];
    float* out = (float*)d_out;

    const int N = in_sizes[0] / FEAT;
    const int E = in_sizes[1] / 2;
    const int K = in_sizes[4];
    const int* src = ei;
    const int* dst = ei + E;

    // Workspace: agg[N*128] | cnt[N] | h1[N*128] | emb[N*128]
    float* agg = (float*)d_ws;
    float* cnt = agg + (size_t)N * FEAT;
    float* h1  = cnt + N;
    float* emb = h1 + (size_t)N * FEAT;

    const int scatterBlocks = 2048;
    const int layerBlocks   = 128;

    // ---- Layer 1 ----
    (void)hipMemsetAsync(agg, 0, (size_t)N * FEAT * sizeof(float), stream);
    (void)hipMemsetAsync(cnt, 0, (size_t)N * sizeof(float), stream);
    scatter_accum_kernel<<<scatterBlocks, 256, 0, stream>>>(x, src, dst, agg, cnt, E, 1);
    sage_layer_wmma<<<layerBlocks, 256, 0, stream>>>(agg, cnt, x, W1l, W1r, b1, h1, N, 1);

    // ---- Layer 2 (reuse cnt: same graph) ----
    (void)hipMemsetAsync(agg, 0, (size_t)N * FEAT * sizeof(float), stream);
    scatter_accum_kernel<<<scatterBlocks, 256, 0, stream>>>(h1, src, dst, agg, cnt, E, 0);
    sage_layer_wmma<<<layerBlocks, 256, 0, stream>>>(agg, cnt, h1, W2l, W2r, b2, emb, N, 0);

    // ---- MLP head ----
    head_kernel<<<1, 256, 0, stream>>>(emb, cur, dstI, nbr, Wl1, bl1, Wl2, bl2, out, K);
}